// GraphConvolutionSageEncoder_48911087567676
// MI455X (gfx1250) — compile-verified
//
#include <hip/hip_runtime.h>
#include <hip/hip_bf16.h>

#define DFEAT 64
#define GN_EPS 1e-5f
#define LK_SLOPE 0.01f
#define NGRAPH 64
#define NLAYER 5
#define RES_START 2

typedef float v2f __attribute__((ext_vector_type(2)));
typedef float v8f __attribute__((ext_vector_type(8)));

// ---------------------------------------------------------------- utilities
__global__ __launch_bounds__(256) void zerof_kernel(float* __restrict__ p, int n) {
    int i = blockIdx.x * 256 + threadIdx.x;
    if (i < n) p[i] = 0.0f;
}

// degree of dst (forward mean) and degree of src (backward mean)
__global__ __launch_bounds__(256) void degree_kernel(const int* __restrict__ ei,
                                                     float* __restrict__ degf,
                                                     float* __restrict__ degb, int E) {
    int e = blockIdx.x * 256 + threadIdx.x;
    if (e < E) {
        atomicAdd(degf + ei[E + e], 1.0f);  // dst degree
        atomicAdd(degb + ei[e], 1.0f);      // src degree
    }
}

__global__ __launch_bounds__(256) void count_batch_kernel(const int* __restrict__ batch,
                                                          float* __restrict__ cntf, int n) {
    int i = blockIdx.x * 256 + threadIdx.x;
    if (i < n) atomicAdd(cntf + batch[i], 1.0f);
}

// batch is sorted -> node range of graph g is [off[g], off[g+1])
__global__ void prefix_off_kernel(const float* __restrict__ cntf, int* __restrict__ off) {
    if (threadIdx.x == 0 && blockIdx.x == 0) {
        int a = 0;
        for (int g = 0; g < NGRAPH; ++g) { off[g] = a; a += (int)(cntf[g] + 0.5f); }
        off[NGRAPH] = a;
    }
}

// ------------------------------------------------- edge scatter (1 wave/edge)
__global__ __launch_bounds__(256) void sage_agg_kernel(const int* __restrict__ ei,
                                                       const float* __restrict__ hin,
                                                       float* __restrict__ aggf,
                                                       float* __restrict__ aggb, int E) {
    int e = (blockIdx.x * 256 + threadIdx.x) >> 5;
    if (e >= E) return;
    int lane = threadIdx.x & 31;
    int src = ei[e];
    int dst = ei[E + e];
    int c = lane * 2;
    const float2 hs = *(const float2*)(hin + (size_t)src * DFEAT + c);
    const float2 hd = *(const float2*)(hin + (size_t)dst * DFEAT + c);
    atomicAdd(aggf + (size_t)dst * DFEAT + c + 0, hs.x);
    atomicAdd(aggf + (size_t)dst * DFEAT + c + 1, hs.y);
    atomicAdd(aggb + (size_t)src * DFEAT + c + 0, hd.x);
    atomicAdd(aggb + (size_t)src * DFEAT + c + 1, hd.y);
}

// -------------------------------------- fused dual GEMM: y = m@Wl + x@Wr + b
// m = 0.5*(aggf/max(degf,1) + aggb/max(degb,1)), computed on the fly.
// One wave computes a 16x64 output stripe via V_WMMA_F32_16X16X4_F32.
// LDS weights are stored K-pair interleaved: sW[kpair*64 + n] = {W[2kp][n], W[2kp+1][n]}
// so each B fragment is one aligned ds_load_b64 (no packing moves).
__global__ __launch_bounds__(256) void sage_gemm_kernel(
    const float* __restrict__ hin, const float* __restrict__ aggf,
    const float* __restrict__ aggb, const float* __restrict__ degf,
    const float* __restrict__ degb, const float* __restrict__ Wl,
    const float* __restrict__ Wr, const float* __restrict__ bvec,
    float* __restrict__ y, int nTiles, int N) {

    __shared__ v2f sWl[(DFEAT / 2) * DFEAT];   // 16 KB, kpair-major
    __shared__ v2f sWr[(DFEAT / 2) * DFEAT];   // 16 KB
    for (int j = threadIdx.x; j < DFEAT * DFEAT; j += 256) {
        int k = j >> 6;          // source row (K)
        int n = j & 63;          // source col (N)
        int di = (k >> 1) * (DFEAT * 2) + n * 2 + (k & 1);
        ((float*)sWl)[di] = Wl[j];
        ((float*)sWr)[di] = Wr[j];
    }
    __syncthreads();

    int wave = (blockIdx.x * 256 + threadIdx.x) >> 5;   // wave-uniform
    if (wave >= nTiles) return;                          // uniform exit, EXEC stays full

    int lane = threadIdx.x & 31;
    int half = lane >> 4;     // half-wave selects K pair per ISA f32 A/B layout
    int lm   = lane & 15;

    int rowRaw = wave * 16 + lm;
    int row = rowRaw < N ? rowRaw : (N - 1);            // clamp (N divisible by 16 anyway)
    float rf = 0.5f / fmaxf(degf[row], 1.0f);           // fold 0.5 combine scale in
    float rb = 0.5f / fmaxf(degb[row], 1.0f);
    const float* hrow = hin  + (size_t)row * DFEAT;
    const float* afr  = aggf + (size_t)row * DFEAT;
    const float* abr  = aggb + (size_t)row * DFEAT;

    v8f acc[4];
#pragma unroll
    for (int ct = 0; ct < 4; ++ct) acc[ct] = v8f{0.f,0.f,0.f,0.f,0.f,0.f,0.f,0.f};

#pragma unroll
    for (int kc = 0; kc < 16; ++kc) {
        int k0 = kc * 4 + half * 2;      // this lane's K base
        int kp = kc * 2 + half;          // K pair index into LDS layout
        v2f am, ax;
        am.x = afr[k0]     * rf + abr[k0]     * rb;
        am.y = afr[k0 + 1] * rf + abr[k0 + 1] * rb;
        ax.x = hrow[k0];
        ax.y = hrow[k0 + 1];
#pragma unroll
        for (int ct = 0; ct < 4; ++ct) {
            int n = ct * 16 + lm;
            v2f bl = sWl[kp * DFEAT + n];   // single ds_load_b64
            v2f br = sWr[kp * DFEAT + n];
            acc[ct] = __builtin_amdgcn_wmma_f32_16x16x4_f32(
                false, am, false, bl, (short)0, acc[ct], false, false);
            acc[ct] = __builtin_amdgcn_wmma_f32_16x16x4_f32(
                false, ax, false, br, (short)0, acc[ct], false, false);
        }
    }

#pragma unroll
    for (int ct = 0; ct < 4; ++ct) {
        int col = ct * 16 + lm;
        float bc = bvec[col];
#pragma unroll
        for (int r = 0; r < 8; ++r) {
            int orow = wave * 16 + half * 8 + r;   // C/D layout: M = 8*half + r, N = lm
            if (orow < N) y[(size_t)orow * DFEAT + col] = acc[ct][r] + bc;
        }
    }
}

// ---------------------------------------- per-graph sum / sumsq (1 block/graph)
__global__ __launch_bounds__(256) void gn_stats_kernel(const float* __restrict__ y,
                                                       const int* __restrict__ off,
                                                       float* __restrict__ sums,
                                                       float* __restrict__ sumsq) {
    int g = blockIdx.x;
    int t = threadIdx.x;
    int d = t & 63;
    int r = off[g] + (t >> 6);
    int end = off[g + 1];
    float s = 0.f, q = 0.f;
    for (; r < end; r += 4) {
        float v = y[(size_t)r * DFEAT + d];
        s += v;
        q += v * v;
    }
    __shared__ float ls[256];
    __shared__ float lq[256];
    ls[t] = s; lq[t] = q;
    __syncthreads();
    if (t < 64) {
        s = ls[t] + ls[t + 64] + ls[t + 128] + ls[t + 192];
        q = lq[t] + lq[t + 64] + lq[t + 128] + lq[t + 192];
        sums[g * DFEAT + t]  = s;
        sumsq[g * DFEAT + t] = q;
    }
}

// ------------------------- GraphNorm + leaky-relu + optional residual, writes h
__global__ __launch_bounds__(256) void gn_apply_kernel(
    const float* __restrict__ y, const int* __restrict__ batch,
    const float* __restrict__ cntf, const float* __restrict__ sums,
    const float* __restrict__ sumsq, const float* __restrict__ w,
    const float* __restrict__ bias, const float* __restrict__ ms,
    float* __restrict__ h, int resflag, int total) {
    int idx = blockIdx.x * 256 + threadIdx.x;
    if (idx >= total) return;
    int n = idx >> 6, d = idx & 63;
    int g = batch[n];
    float c    = fmaxf(cntf[g], 1.0f);
    float mean = sums[g * DFEAT + d] / c;
    float msq  = sumsq[g * DFEAT + d] / c;
    float m    = ms[d];
    float var  = msq - (2.0f * m - m * m) * mean * mean;   // E[(y-ms*mean)^2]
    float o = y[idx] - m * mean;
    o *= rsqrtf(var + GN_EPS);
    o = w[d] * o + bias[d];
    o = (o >= 0.0f) ? o : LK_SLOPE * o;
    h[idx] = resflag ? (h[idx] + o) : o;
}

// -------------------------------------------------- global mean pool -> d_out
__global__ __launch_bounds__(256) void pool_kernel(const float* __restrict__ h,
                                                   const int* __restrict__ off,
                                                   const float* __restrict__ cntf,
                                                   float* __restrict__ out) {
    int g = blockIdx.x;
    int t = threadIdx.x;
    int d = t & 63;
    int r = off[g] + (t >> 6);
    int end = off[g + 1];
    float s = 0.f;
    for (; r < end; r += 4) s += h[(size_t)r * DFEAT + d];
    __shared__ float ls[256];
    ls[t] = s;
    __syncthreads();
    if (t < 64) {
        s = ls[t] + ls[t + 64] + ls[t + 128] + ls[t + 192];
        out[g * DFEAT + t] = s / fmaxf(cntf[g], 1.0f);
    }
}

// ---------------------------------------------------------------------- host
extern "C" void kernel_launch(void* const* d_in, const int* in_sizes, int n_in,
                              void* d_out, int out_size, void* d_ws, size_t ws_size,
                              hipStream_t stream) {
    const float* x     = (const float*)d_in[0];
    const int*   ei    = (const int*)d_in[1];
    const int*   batch = (const int*)d_in[2];
    const float* W_l   = (const float*)d_in[3];
    const float* W_r   = (const float*)d_in[4];
    const float* bvec  = (const float*)d_in[5];
    const float* gn_w  = (const float*)d_in[6];
    const float* gn_b  = (const float*)d_in[7];
    const float* gn_ms = (const float*)d_in[8];
    float* out = (float*)d_out;

    const int N = in_sizes[0] / DFEAT;     // 100000
    const int E = in_sizes[1] / 2;         // 1600000

    // workspace carve-out
    char* ws = (char*)d_ws;
    size_t ofs = 0;
    auto carve = [&](size_t bytes) -> void* {
        void* p = ws + ofs;
        ofs += (bytes + 255) & ~(size_t)255;
        return p;
    };
    float* h    = (float*)carve((size_t)N * DFEAT * 4);
    float* y    = (float*)carve((size_t)N * DFEAT * 4);
    float* aggf = (float*)carve((size_t)2 * N * DFEAT * 4);  // aggf + aggb contiguous
    float* aggb = aggf + (size_t)N * DFEAT;
    float* degf  = (float*)carve((size_t)N * 4);
    float* degb  = (float*)carve((size_t)N * 4);
    float* cntf  = (float*)carve(NGRAPH * 4);
    int*   off   = (int*)carve((NGRAPH + 1) * 4);
    float* sums  = (float*)carve(NGRAPH * DFEAT * 4);
    float* sumsq = (float*)carve(NGRAPH * DFEAT * 4);

    const int total = N * DFEAT;
    const int nTiles = (N + 15) / 16;

    // one-time (per launch) precompute: degrees, graph counts, graph offsets
    zerof_kernel<<<(N + 255) / 256, 256, 0, stream>>>(degf, N);
    zerof_kernel<<<(N + 255) / 256, 256, 0, stream>>>(degb, N);
    zerof_kernel<<<1, 256, 0, stream>>>(cntf, NGRAPH);
    degree_kernel<<<(E + 255) / 256, 256, 0, stream>>>(ei, degf, degb, E);
    count_batch_kernel<<<(N + 255) / 256, 256, 0, stream>>>(batch, cntf, N);
    prefix_off_kernel<<<1, 32, 0, stream>>>(cntf, off);

    const float* hin = x;
    for (int i = 0; i < NLAYER; ++i) {
        zerof_kernel<<<(2 * total + 255) / 256, 256, 0, stream>>>(aggf, 2 * total);
        sage_agg_kernel<<<(E * 32 + 255) / 256, 256, 0, stream>>>(ei, hin, aggf, aggb, E);
        sage_gemm_kernel<<<(nTiles * 32 + 255) / 256, 256, 0, stream>>>(
            hin, aggf, aggb, degf, degb,
            W_l + (size_t)i * DFEAT * DFEAT, W_r + (size_t)i * DFEAT * DFEAT,
            bvec + (size_t)i * DFEAT, y, nTiles, N);
        gn_stats_kernel<<<NGRAPH, 256, 0, stream>>>(y, off, sums, sumsq);
        gn_apply_kernel<<<(total + 255) / 256, 256, 0, stream>>>(
            y, batch, cntf, sums, sumsq,
            gn_w + (size_t)i * DFEAT, gn_b + (size_t)i * DFEAT, gn_ms + (size_t)i * DFEAT,
            h, (i >= RES_START) ? 1 : 0, total);
        hin = h;
    }

    pool_kernel<<<NGRAPH, 256, 0, stream>>>(h, off, cntf, out);
}